// Model_90280212562314
// MI455X (gfx1250) — compile-verified
//
#include <hip/hip_runtime.h>
#include <hip/hip_bf16.h>

// ---------------- problem constants ----------------
static constexpr int kB   = 16;
static constexpr int kT   = 1024;
static constexpr int kN   = 128;   // nodes
static constexpr int kWIN = 30;
static constexpr int kNW  = 498;   // number of windows: arange(0, 995, 2)
static constexpr int kD   = 256;   // d_model
static constexpr int kIN  = 384;   // 3 * N
static constexpr int kTS  = 34;    // transposed-LDS stride (even + conflict-free)

typedef __attribute__((ext_vector_type(2))) float v2f;
typedef __attribute__((ext_vector_type(8))) float v8f;

// f32 WMMA: D(16x16) = A(16x4) * B(4x16) + C   (wave32, A/B = 2 VGPRs, C/D = 8 VGPRs)
__device__ __forceinline__ v8f wmma_f32_k4(v2f a, v2f b, v8f c) {
    return __builtin_amdgcn_wmma_f32_16x16x4_f32(
        /*neg_a=*/false, a, /*neg_b=*/false, b,
        /*c_mod=*/(short)0, c, /*reuse_a=*/false, /*reuse_b=*/false);
}

// Async DMA of one 30x128 f32 window (3840 floats) global -> LDS.
// Exactly 15 b32 async ops per thread (uniform, no divergence), so every wave
// issues exactly 15 ASYNCcnt increments per window.
__device__ __forceinline__ void async_copy_window(const float* __restrict__ src,
                                                  float* dstLds, int tid) {
    // Low 32 bits of a generic LDS pointer are the wave-relative LDS byte addr.
    const uint32_t lds0 = (uint32_t)(uintptr_t)dstLds;
#pragma unroll
    for (int i = 0; i < 15; ++i) {
        const int idx = tid + i * 256;               // 0 .. 3839
        const uint64_t ga = (uint64_t)(uintptr_t)(src + idx);
        const uint32_t la = lds0 + (uint32_t)idx * 4u;
        asm volatile("global_load_async_to_lds_b32 %0, %1, off"
                     :: "v"(la), "v"(ga) : "memory");
    }
}

// ============================================================================
// Pass 1 / Pass 2 over sliding windows.
//  grid = (8 row-tiles, B).  256 threads = 8 waves; wave w owns column tile w.
//  Double-buffered async global->LDS staging of the raw window overlaps the
//  centering + WMMA of the previous window.
//  PASS==1: accumulate sum of per-window correlation -> mean_lo
//  PASS==2: accumulate (corr - mean_lo)^{2,4,8}      -> cm2/cm4/cm8
// ============================================================================
template <int PASS>
__global__ __launch_bounds__(256)
void win_kernel(const float* __restrict__ x,
                float* __restrict__ mean_lo,
                float* __restrict__ cm_all /* 3 x B x N x N */)
{
    const int mi   = blockIdx.x;          // row tile 0..7
    const int b    = blockIdx.y;
    const int tid  = threadIdx.x;
    const int wid  = tid >> 5;            // wave id = column tile
    const int lane = tid & 31;
    const int half = lane >> 4;
    const int l16  = lane & 15;
    const int ni   = wid;

    __shared__ float raw[2][kWIN * kN];   // raw window double buffer (2 x 15 KB)
    __shared__ float xsT[kN * kTS];       // centered window, TRANSPOSED [col][k0..31]
    __shared__ float rstd[kN];

    const float inv29 = 1.0f / (29.0f + 1e-8f);
    const int gmBase  = mi * 16 + half * 8;
    const int gn      = ni * 16 + l16;    // global column of this lane
    const int mrow    = mi * 16 + l16;    // global row used for A-fragment

    v8f sum_corr = {};
    v8f s2 = {}, s4 = {}, s8 = {};
    float meanv[8];
    if constexpr (PASS == 2) {
        for (int r = 0; r < 8; ++r)
            meanv[r] = mean_lo[((long)b * kN + (gmBase + r)) * kN + gn];
    }

    const float* xb = x + (long)b * kT * kN;

    // prologue: kick off window 0
    async_copy_window(xb, &raw[0][0], tid);

    for (int w = 0; w < kNW; ++w) {
        const int buf = w & 1;
        // start DMA for the next window into the other buffer
        if (w + 1 < kNW) {
            async_copy_window(xb + (long)(2 * (w + 1)) * kN, &raw[buf ^ 1][0], tid);
            // <=15 outstanding left => all of window w's 15 ops have landed
            asm volatile("s_wait_asynccnt 0xf" ::: "memory");
        } else {
            asm volatile("s_wait_asynccnt 0x0" ::: "memory");
        }
        __syncthreads();                          // all waves' portions of raw[buf] ready

        if (tid < kN) {                           // center columns + rstd, write transposed
            const int c = tid;
            float mean = 0.0f;
            for (int l = 0; l < kWIN; ++l) mean += raw[buf][l * kN + c];
            mean *= (1.0f / kWIN);
            float ss = 0.0f;
            float* xc = &xsT[c * kTS];
            for (int l = 0; l < kWIN; ++l) {
                const float v = raw[buf][l * kN + c] - mean;
                xc[l] = v;
                ss += v * v;
            }
            xc[30] = 0.0f;                        // zero K padding
            xc[31] = 0.0f;
            const float var = fmaxf(ss * inv29, 1e-8f);
            rstd[c] = rsqrtf(var);
        }
        __syncthreads();

        // cov tile: (16x32) * (32x16) via 8 x K4 WMMA; fragments are single
        // aligned ds_load_b64 (stride kTS=34 keeps 8B alignment, no bank conflicts)
        v8f acc = {};
        const float* am = &xsT[mrow * kTS];
        const float* bn = &xsT[gn * kTS];
#pragma unroll
        for (int k0 = 0; k0 < 32; k0 += 4) {
            const int ka = k0 + 2 * half;         // even
            const v2f a  = *(const v2f*)(am + ka);
            const v2f bb = *(const v2f*)(bn + ka);
            acc = wmma_f32_k4(a, bb, acc);
        }

        const float rn = rstd[gn];
        for (int r = 0; r < 8; ++r) {
            const int gm = gmBase + r;
            float corr = acc[r] * inv29 * rstd[gm] * rn;
            if (gm == gn) corr = 1.0f;            // mask diag
            if constexpr (PASS == 1) {
                sum_corr[r] += corr;
            } else {
                const float c  = corr - meanv[r];
                const float c2 = c * c;  s2[r] += c2;
                const float c4 = c2 * c2; s4[r] += c4;
                s8[r] += c4 * c4;
            }
        }
        __syncthreads();                          // xsT reused next iteration
    }

    const float invW = 1.0f / (float)kNW;
    if constexpr (PASS == 1) {
        for (int r = 0; r < 8; ++r)
            mean_lo[((long)b * kN + gmBase + r) * kN + gn] = sum_corr[r] * invW;
    } else {
        const long base = (long)b * kN * kN;
        const long bs   = (long)kB * kN * kN;
        for (int r = 0; r < 8; ++r) {
            const long o = base + (long)(gmBase + r) * kN + gn;
            cm_all[0 * bs + o] = s2[r] * invW;
            cm_all[1 * bs + o] = s4[r] * invW;
            cm_all[2 * bs + o] = s8[r] * invW;
        }
    }
}

// ============================================================================
// Prep: per-moment row centering (Xc, den for rowwise_corr) + 384-wide
// layer-norm of concatenated moments -> X.  grid = (B), 256 threads.
// ============================================================================
__global__ void prep_kernel(const float* __restrict__ cm_all,
                            const float* __restrict__ gamma,
                            const float* __restrict__ beta,
                            float* __restrict__ Xc_all,
                            float* __restrict__ den_all,
                            float* __restrict__ X)
{
    const int b = blockIdx.x;
    const int t = threadIdx.x;
    if (t >= kN) return;
    const int n = t;
    const long bs = (long)kB * kN * kN;

    for (int d = 0; d < 3; ++d) {
        const float* cm = cm_all + d * bs + (long)b * kN * kN + (long)n * kN;
        float mean = 0.0f;
        for (int f = 0; f < kN; ++f) mean += cm[f];
        mean *= (1.0f / kN);
        float* xc = Xc_all + d * bs + (long)b * kN * kN + (long)n * kN;
        float ss = 0.0f;
        for (int f = 0; f < kN; ++f) {
            const float v = cm[f] - mean;
            xc[f] = v;
            ss += v * v;
        }
        den_all[((long)d * kB + b) * kN + n] = sqrtf(fmaxf(ss, 1e-8f));
    }

    // layer norm over the 384 concatenated features of row n
    const float* c2 = cm_all + 0 * bs + (long)b * kN * kN + (long)n * kN;
    const float* c4 = cm_all + 1 * bs + (long)b * kN * kN + (long)n * kN;
    const float* c8 = cm_all + 2 * bs + (long)b * kN * kN + (long)n * kN;
    float mu = 0.0f;
    for (int f = 0; f < kN; ++f) mu += c2[f] + c4[f] + c8[f];
    mu *= (1.0f / kIN);
    float var = 0.0f;
    for (int f = 0; f < kN; ++f) {
        const float a0 = c2[f] - mu, a1 = c4[f] - mu, a2 = c8[f] - mu;
        var += a0 * a0 + a1 * a1 + a2 * a2;
    }
    var *= (1.0f / kIN);
    const float inv = rsqrtf(var + 1e-5f);
    float* Xr = X + ((long)b * kN + n) * kIN;
    for (int f = 0; f < kN; ++f) Xr[f]            = (c2[f] - mu) * inv * gamma[f]            + beta[f];
    for (int f = 0; f < kN; ++f) Xr[kN + f]       = (c4[f] - mu) * inv * gamma[kN + f]       + beta[kN + f];
    for (int f = 0; f < kN; ++f) Xr[2 * kN + f]   = (c8[f] - mu) * inv * gamma[2 * kN + f]   + beta[2 * kN + f];
}

// ============================================================================
// ho = rowwise_corr(cm_d) for d in {2,4,8}; Asum = mean of the three.
// num = Xc * Xc^T via f32 WMMA.  grid = (8 row-tiles, B), 8 waves/block.
// ============================================================================
__global__ void ho_kernel(const float* __restrict__ Xc_all,
                          const float* __restrict__ den_all,
                          float* __restrict__ Asum)
{
    const int mi   = blockIdx.x;
    const int b    = blockIdx.y;
    const int wid  = threadIdx.x >> 5;
    const int lane = threadIdx.x & 31;
    const int half = lane >> 4;
    const int l16  = lane & 15;
    const int n    = wid * 16 + l16;      // global column
    const int m    = mi * 16 + l16;       // row for A fragment
    const long bs  = (long)kB * kN * kN;

    v8f A_acc = {};
    for (int d = 0; d < 3; ++d) {
        const float* Xc  = Xc_all  + d * bs + (long)b * kN * kN;
        const float* den = den_all + ((long)d * kB + b) * kN;
        v8f acc = {};
        for (int k0 = 0; k0 < kN; k0 += 4) {
            const int ka = k0 + 2 * half;
            v2f a, bb;
            a.x  = Xc[(long)m * kN + ka];  a.y  = Xc[(long)m * kN + ka + 1];
            bb.x = Xc[(long)n * kN + ka];  bb.y = Xc[(long)n * kN + ka + 1]; // B(k,n) = Xc[n][k]
            acc = wmma_f32_k4(a, bb, acc);
        }
        const float dn = den[n];
        for (int r = 0; r < 8; ++r) {
            const int gm = mi * 16 + half * 8 + r;
            float c = acc[r] / (den[gm] * dn);
            if (gm == n) c = 1.0f;
            A_acc[r] += c;
        }
    }
    float* Ab = Asum + (long)b * kN * kN;
    for (int r = 0; r < 8; ++r) {
        const int gm = mi * 16 + half * 8 + r;
        Ab[(long)gm * kN + n] = A_acc[r] * (1.0f / 3.0f);
    }
}

// ============================================================================
// Symmetrize + add identity + sym-normalize.  grid = (B), 256 threads.
// ============================================================================
__global__ void normadj_kernel(const float* __restrict__ Asum,
                               float* __restrict__ Anorm)
{
    const int b = blockIdx.x;
    const int t = threadIdx.x;
    const float* A = Asum + (long)b * kN * kN;
    float* An = Anorm + (long)b * kN * kN;
    __shared__ float dis[kN];
    if (t < kN) {
        const int i = t;
        float deg = 0.0f;
        for (int j = 0; j < kN; ++j) {
            float v = 0.5f * (A[i * kN + j] + A[j * kN + i]) + (i == j ? 1.0f : 0.0f);
            deg += v;
        }
        dis[i] = rsqrtf(fmaxf(deg, 1e-8f));
    }
    __syncthreads();
    for (int e = t; e < kN * kN; e += 256) {
        const int i = e >> 7, j = e & 127;
        const float v = 0.5f * (A[i * kN + j] + A[j * kN + i]) + (i == j ? 1.0f : 0.0f);
        An[e] = v * dis[i] * dis[j];
    }
}

// ============================================================================
// Generic batched f32 WMMA GEMM: C[b] = act(P[b] @ Q[b] + bias)
//  P: [b][128][K]   (pStride = per-batch elements)
//  Q: [b][K][Npar]  (qStride = 0 => shared weights)
//  grid = (8 row-tiles, B), 256 threads; waves stride the column tiles.
// ============================================================================
__global__ void gemm_wmma(const float* __restrict__ P, long pStride,
                          const float* __restrict__ Q, long qStride,
                          const float* __restrict__ bias,
                          float* __restrict__ C, long cStride,
                          int K, int Npar, int relu)
{
    const int mi   = blockIdx.x;
    const int b    = blockIdx.y;
    const int wid  = threadIdx.x >> 5;
    const int lane = threadIdx.x & 31;
    const int half = lane >> 4;
    const int l16  = lane & 15;
    const float* Pb = P + (long)b * pStride;
    const float* Qb = Q + (long)b * qStride;
    float* Cb = C + (long)b * cStride;
    const int m = mi * 16 + l16;
    const int nt = Npar >> 4;

    for (int ni = wid; ni < nt; ni += 8) {
        const int n = ni * 16 + l16;
        v8f acc = {};
        for (int k0 = 0; k0 < K; k0 += 4) {
            const int ka = k0 + 2 * half;
            v2f a, bb;
            a.x  = Pb[(long)m * K + ka];       a.y  = Pb[(long)m * K + ka + 1];
            bb.x = Qb[(long)ka * Npar + n];    bb.y = Qb[(long)(ka + 1) * Npar + n];
            acc = wmma_f32_k4(a, bb, acc);
        }
        const float bv = bias ? bias[n] : 0.0f;
        for (int r = 0; r < 8; ++r) {
            const int gm = mi * 16 + half * 8 + r;
            float v = acc[r] + bv;
            if (relu) v = fmaxf(v, 0.0f);
            Cb[(long)gm * Npar + n] = v;
        }
    }
}

// ============================================================================
// Head: g = mean over nodes of H2, out = sigmoid(g @ Wc + bc).
// grid = (B), 256 threads (one per d_model channel).
// ============================================================================
__global__ void head_kernel(const float* __restrict__ H2,
                            const float* __restrict__ Wc,
                            const float* __restrict__ bc,
                            float* __restrict__ out)
{
    const int b = blockIdx.x;
    const int t = threadIdx.x;
    const float* Hb = H2 + (long)b * kN * kD;
    float s = 0.0f;
    for (int r = 0; r < kN; ++r) s += Hb[r * kD + t];
    const float g = s * (1.0f / kN);
    __shared__ float red[256];
    red[t] = g * Wc[t];
    __syncthreads();
    for (int off = 128; off > 0; off >>= 1) {
        if (t < off) red[t] += red[t + off];
        __syncthreads();
    }
    if (t == 0) out[b] = 1.0f / (1.0f + __expf(-(red[0] + bc[0])));
}

// ============================================================================
// Host launcher
// ============================================================================
extern "C" void kernel_launch(void* const* d_in, const int* in_sizes, int n_in,
                              void* d_out, int out_size, void* d_ws, size_t ws_size,
                              hipStream_t stream)
{
    const float* x     = (const float*)d_in[0];
    const float* gam   = (const float*)d_in[1];
    const float* bet   = (const float*)d_in[2];
    const float* W1    = (const float*)d_in[3];
    const float* b1    = (const float*)d_in[4];
    const float* W2    = (const float*)d_in[5];
    const float* b2    = (const float*)d_in[6];
    const float* Wc    = (const float*)d_in[7];
    const float* bc    = (const float*)d_in[8];
    float* out = (float*)d_out;

    // workspace layout (floats)
    float* ws = (float*)d_ws;
    const long NN   = (long)kN * kN;
    float* mean_lo = ws;                              // B*N*N
    float* cm      = mean_lo + kB * NN;               // 3*B*N*N
    float* Xc      = cm      + 3 * kB * NN;           // 3*B*N*N
    float* den     = Xc      + 3 * kB * NN;           // 3*B*N
    float* Asum    = den     + 3 * (long)kB * kN;     // B*N*N
    float* Anorm   = Asum    + kB * NN;               // B*N*N
    float* X       = Anorm   + kB * NN;               // B*N*384
    float* XW      = X       + (long)kB * kN * kIN;   // B*N*256
    float* H1      = XW      + (long)kB * kN * kD;    // B*N*256
    float* HW      = H1      + (long)kB * kN * kD;    // B*N*256
    float* H2      = HW      + (long)kB * kN * kD;    // B*N*256

    const dim3 gridT(8, kB);

    // Pass 1: mean of per-window correlation matrices
    win_kernel<1><<<gridT, 256, 0, stream>>>(x, mean_lo, cm);
    // Pass 2: centered moments (2,4,8)
    win_kernel<2><<<gridT, 256, 0, stream>>>(x, mean_lo, cm);
    // Row-centering + dens + layer-norm features
    prep_kernel<<<kB, 256, 0, stream>>>(cm, gam, bet, Xc, den, X);
    // Higher-order adjacency (rowwise corr via WMMA)
    ho_kernel<<<gridT, 256, 0, stream>>>(Xc, den, Asum);
    // Symmetrize + normalize adjacency
    normadj_kernel<<<kB, 256, 0, stream>>>(Asum, Anorm);
    // GCN layer 1: XW = X@W1 + b1 ; H1 = relu(Anorm @ XW)
    gemm_wmma<<<gridT, 256, 0, stream>>>(X, (long)kN * kIN, W1, 0, b1,
                                         XW, (long)kN * kD, kIN, kD, 0);
    gemm_wmma<<<gridT, 256, 0, stream>>>(Anorm, NN, XW, (long)kN * kD, nullptr,
                                         H1, (long)kN * kD, kN, kD, 1);
    // GCN layer 2: HW = H1@W2 + b2 ; H2 = relu(Anorm @ HW)
    gemm_wmma<<<gridT, 256, 0, stream>>>(H1, (long)kN * kD, W2, 0, b2,
                                         HW, (long)kN * kD, kD, kD, 0);
    gemm_wmma<<<gridT, 256, 0, stream>>>(Anorm, NN, HW, (long)kN * kD, nullptr,
                                         H2, (long)kN * kD, kN, kD, 1);
    // Pool + sigmoid head
    head_kernel<<<kB, 256, 0, stream>>>(H2, Wc, bc, out);
}